// GraphSAGE_26431228739933
// MI455X (gfx1250) — compile-verified
//
#include <hip/hip_runtime.h>

typedef float v2f __attribute__((ext_vector_type(2)));
typedef float v8f __attribute__((ext_vector_type(8)));

#define IN_DIM 6
#define HDIM 64

// ---------------- zero scratch ----------------
__global__ void sage_zero_kernel(float* __restrict__ p, int n) {
    int i = blockIdx.x * blockDim.x + threadIdx.x;
    if (i < n) p[i] = 0.0f;
}

// ---------------- layer-1 scatter: degree count + 6-dim feature sum ----------------
__global__ void sage_scatter1_kernel(const float* __restrict__ x,
                                     const long long* __restrict__ src,
                                     const long long* __restrict__ dst,
                                     float* __restrict__ cnt,
                                     float* __restrict__ agg1, int E) {
    int e = blockIdx.x * blockDim.x + threadIdx.x;
    if (e >= E) return;
    int s = (int)src[e];
    int d = (int)dst[e];
    atomicAdd(&cnt[d], 1.0f);
#pragma unroll
    for (int k = 0; k < IN_DIM; ++k)
        atomicAdd(&agg1[d * IN_DIM + k], x[s * IN_DIM + k]);
}

// ---------------- layer-1 linear (K=6, scalar VALU) + ReLU ----------------
__global__ void sage_layer1_kernel(const float* __restrict__ x,
                                   const float* __restrict__ agg1,
                                   const float* __restrict__ cnt,
                                   const float* __restrict__ W1l,
                                   const float* __restrict__ b1,
                                   const float* __restrict__ W1r,
                                   float* __restrict__ h, int N) {
    int i = blockIdx.x * blockDim.x + threadIdx.x;
    if (i >= N * HDIM) return;
    int n = i >> 6;
    int c = i & 63;
    float inv = 1.0f / fmaxf(cnt[n], 1.0f);
    float acc = b1[c];
#pragma unroll
    for (int k = 0; k < IN_DIM; ++k) {
        acc = fmaf(agg1[n * IN_DIM + k] * inv, W1l[k * HDIM + c], acc);
        acc = fmaf(x[n * IN_DIM + k],          W1r[k * HDIM + c], acc);
    }
    h[i] = fmaxf(acc, 0.0f);
}

// ---------------- layer-2 scatter: 64-dim, float4 gathers + atomics ----------------
__global__ void sage_scatter2_kernel(const float* __restrict__ h,
                                     const long long* __restrict__ src,
                                     const long long* __restrict__ dst,
                                     float* __restrict__ agg2, int E) {
    int t = blockIdx.x * blockDim.x + threadIdx.x;
    int e = t >> 4;                 // 16 threads per edge, 4 floats each
    if (e >= E) return;
    int part = (t & 15) << 2;
    int s = (int)src[e];
    int d = (int)dst[e];
    const float4 v = *reinterpret_cast<const float4*>(&h[s * HDIM + part]);
    float* p = &agg2[d * HDIM + part];
    atomicAdd(p + 0, v.x);
    atomicAdd(p + 1, v.y);
    atomicAdd(p + 2, v.z);
    atomicAdd(p + 3, v.w);
}

// ---------------- layer-2 GEMM: fp32 WMMA 16x16x4 ----------------
// out[16 rows x 64 cols] per block; wave w handles column tile w (n0 = 16*w).
// C = bias; C += mean2(16xK) @ W2_l(Kx16); C += h(16xK) @ W2_r(Kx16).
__global__ void __launch_bounds__(128)
sage_layer2_wmma_kernel(const float* __restrict__ h,
                        const float* __restrict__ agg2,
                        const float* __restrict__ cnt,
                        const float* __restrict__ W2l,
                        const float* __restrict__ b2,
                        const float* __restrict__ W2r,
                        float* __restrict__ out) {
    const int lane  = threadIdx.x & 31;
    const int wave  = threadIdx.x >> 5;        // 0..3 -> output column tile
    const int m0    = blockIdx.x * 16;         // output row tile
    const int n0    = wave * 16;

    const int halfsel = lane >> 4;             // 0: lanes 0-15, 1: lanes 16-31
    const int mrow  = m0 + (lane & 15);        // A-matrix: M = lane (both halves)
    const int ncol  = n0 + (lane & 15);        // B/C/D: N = lane (both halves)
    const int khalf = halfsel << 1;            // K offset within a 4-step: 0 or 2

    const float invc = 1.0f / fmaxf(cnt[mrow], 1.0f);

    // C/D init: bias is per-column, identical across the 8 row-VGPRs
    v8f c;
    const float bias = b2[ncol];
#pragma unroll
    for (int i = 0; i < 8; ++i) c[i] = bias;

    // ---- mean2 @ W2_l ----
#pragma unroll
    for (int kk = 0; kk < HDIM; kk += 4) {
        const int ka = kk + khalf;
        v2f a, b;
        a[0] = agg2[mrow * HDIM + ka]     * invc;   // A[M=mrow][K=ka]
        a[1] = agg2[mrow * HDIM + ka + 1] * invc;   // A[M=mrow][K=ka+1]
        b[0] = W2l[ka * HDIM + ncol];               // B[K=ka][N=ncol]
        b[1] = W2l[(ka + 1) * HDIM + ncol];         // B[K=ka+1][N=ncol]
        c = __builtin_amdgcn_wmma_f32_16x16x4_f32(
                false, a, false, b, (short)0, c, false, false);
    }

    // ---- h @ W2_r ----
#pragma unroll
    for (int kk = 0; kk < HDIM; kk += 4) {
        const int ka = kk + khalf;
        v2f a, b;
        a[0] = h[mrow * HDIM + ka];
        a[1] = h[mrow * HDIM + ka + 1];
        b[0] = W2r[ka * HDIM + ncol];
        b[1] = W2r[(ka + 1) * HDIM + ncol];
        c = __builtin_amdgcn_wmma_f32_16x16x4_f32(
                false, a, false, b, (short)0, c, false, false);
    }

    // D layout: VGPR i -> M = i + 8*halfsel, N = ncol
#pragma unroll
    for (int i = 0; i < 8; ++i) {
        const int row = m0 + halfsel * 8 + i;
        out[row * HDIM + ncol] = c[i];
    }
}

extern "C" void kernel_launch(void* const* d_in, const int* in_sizes, int n_in,
                              void* d_out, int out_size, void* d_ws, size_t ws_size,
                              hipStream_t stream) {
    const float*     x   = (const float*)d_in[0];
    const long long* ei  = (const long long*)d_in[1];   // int64 edge_index [2, E]
    const float*     W1l = (const float*)d_in[2];
    const float*     b1  = (const float*)d_in[3];
    const float*     W1r = (const float*)d_in[4];
    const float*     W2l = (const float*)d_in[5];
    const float*     b2  = (const float*)d_in[6];
    const float*     W2r = (const float*)d_in[7];
    float* out = (float*)d_out;

    const int N = in_sizes[0] / IN_DIM;   // 50000 (divisible by 16)
    const int E = in_sizes[1] / 2;        // 1600000
    const long long* src = ei;            // row 0
    const long long* dst = ei + E;        // row 1

    // Workspace layout (floats): cnt[N] | agg1[N*6] | agg2[N*64] | h[N*64]
    float* ws   = (float*)d_ws;
    float* cnt  = ws;
    float* agg1 = cnt + N;
    float* agg2 = agg1 + (size_t)N * IN_DIM;
    float* hbuf = agg2 + (size_t)N * HDIM;

    // Zero only the atomic accumulators (h is fully overwritten each call)
    const int zcount = N + N * IN_DIM + N * HDIM;
    sage_zero_kernel<<<(zcount + 255) / 256, 256, 0, stream>>>(ws, zcount);

    sage_scatter1_kernel<<<(E + 255) / 256, 256, 0, stream>>>(x, src, dst, cnt, agg1, E);

    sage_layer1_kernel<<<(N * HDIM + 255) / 256, 256, 0, stream>>>(
        x, agg1, cnt, W1l, b1, W1r, hbuf, N);

    sage_scatter2_kernel<<<((E * 16) + 255) / 256, 256, 0, stream>>>(
        hbuf, src, dst, agg2, E);

    sage_layer2_wmma_kernel<<<N / 16, 128, 0, stream>>>(
        hbuf, agg2, cnt, W2l, b2, W2r, out);
}